// GatedMultiScalePooling_70514773066108
// MI455X (gfx1250) — compile-verified
//
#include <hip/hip_runtime.h>
#include <math.h>

#define B_ 32
#define N_ 4096
#define D_ 512
#define H_ 128
#define OUT_ 512
#define NCHUNK_ 16
#define CHUNK_ 256   // N_ / NCHUNK_

typedef __attribute__((ext_vector_type(2))) float v2f;
typedef __attribute__((ext_vector_type(8))) float v8f;

// Branch-free tanh: exact at +/-inf, ~1e-7 abs error, single v_exp_f32.
__device__ __forceinline__ float fast_tanh(float v) {
  const float e = __expf(2.0f * v);
  return 1.0f - 2.0f / (e + 1.0f);
}

// ---------------------------------------------------------------------------
// Kernel 1: scores[b*N+n] = tanh(x[row,:] @ W1 + b1) @ w2 + b2  via f32 WMMA.
// Block = 256 threads = 8 waves; each wave owns 32 rows (two 16-row A tiles
// sharing one B load), block covers 256 rows. Grid = B*N/256 = 512 blocks.
// ---------------------------------------------------------------------------
__global__ __launch_bounds__(256) void attn_score_kernel(
    const float* __restrict__ x, const float* __restrict__ w1,
    const float* __restrict__ b1, const float* __restrict__ w2,
    const float* __restrict__ b2, float* __restrict__ scores)
{
  const int wave = threadIdx.x >> 5;
  const int lane = threadIdx.x & 31;
  const int half = lane >> 4;    // 0: lanes 0-15, 1: lanes 16-31
  const int sub  = lane & 15;

  const size_t row0 = (size_t)blockIdx.x * 256 + (size_t)wave * 32;
  const size_t row1 = row0 + 16;

  v8f c0[8] = {};
  v8f c1[8] = {};

  const float* xr0 = x + (row0 + sub) * D_;
  const float* xr1 = x + (row1 + sub) * D_;

  for (int k = 0; k < D_; k += 4) {
    const int krow = k + 2 * half;           // K-pair owned by this lane-half
    v2f a0 = __builtin_nontemporal_load((const v2f*)(xr0 + krow));
    v2f a1 = __builtin_nontemporal_load((const v2f*)(xr1 + krow));
#pragma unroll
    for (int t = 0; t < 8; ++t) {            // 8 column tiles of h (128 = 8*16)
      const int col = t * 16 + sub;
      const float* wp = w1 + (size_t)krow * H_ + col;
      v2f bv;
      bv.x = wp[0];                          // W1[krow][col]
      bv.y = wp[H_];                         // W1[krow+1][col]
      c0[t] = __builtin_amdgcn_wmma_f32_16x16x4_f32(
          false, a0, false, bv, (short)0, c0[t], false, false);
      c1[t] = __builtin_amdgcn_wmma_f32_16x16x4_f32(
          false, a1, false, bv, (short)0, c1[t], false, false);
    }
  }

  // Epilogue: score[row] = sum_col tanh(h + b1[col]) * w2[col]
  float rs0[8], rs1[8];
#pragma unroll
  for (int r = 0; r < 8; ++r) { rs0[r] = 0.0f; rs1[r] = 0.0f; }
#pragma unroll
  for (int t = 0; t < 8; ++t) {
    const int col = t * 16 + sub;
    const float bb = b1[col];
    const float ww = w2[col];
#pragma unroll
    for (int r = 0; r < 8; ++r) {
      rs0[r] += fast_tanh(c0[t][r] + bb) * ww;
      rs1[r] += fast_tanh(c1[t][r] + bb) * ww;
    }
  }
  // Per-row reduce across the 16 lanes of each half (C: row = r + 8*half).
#pragma unroll
  for (int r = 0; r < 8; ++r) {
#pragma unroll
    for (int m = 8; m >= 1; m >>= 1) {
      rs0[r] += __shfl_xor(rs0[r], m, 16);
      rs1[r] += __shfl_xor(rs1[r], m, 16);
    }
  }
  if (sub == 0) {
    const float beta = b2[0];
#pragma unroll
    for (int r = 0; r < 8; ++r) {
      scores[row0 + (size_t)(half * 8 + r)] = rs0[r] + beta;
      scores[row1 + (size_t)(half * 8 + r)] = rs1[r] + beta;
    }
  }
}

// ---------------------------------------------------------------------------
// Block-wide reductions (256 threads).
// ---------------------------------------------------------------------------
__device__ __forceinline__ float block_reduce_sum(float v, float* red, int tid) {
  red[tid] = v; __syncthreads();
#pragma unroll
  for (int s = 128; s > 0; s >>= 1) {
    if (tid < s) red[tid] += red[tid + s];
    __syncthreads();
  }
  float r = red[0]; __syncthreads();
  return r;
}

__device__ __forceinline__ float block_reduce_max(float v, float* red, int tid) {
  red[tid] = v; __syncthreads();
#pragma unroll
  for (int s = 128; s > 0; s >>= 1) {
    if (tid < s) red[tid] = fmaxf(red[tid], red[tid + s]);
    __syncthreads();
  }
  float r = red[0]; __syncthreads();
  return r;
}

// ---------------------------------------------------------------------------
// Kernel 2: masked softmax over N per batch + valid count. One block per b.
// ---------------------------------------------------------------------------
__global__ __launch_bounds__(256) void softmax_mask_kernel(
    const float* __restrict__ scores, const int* __restrict__ mask,
    float* __restrict__ weights, float* __restrict__ inv_valid)
{
  __shared__ float red[256];
  const int b = blockIdx.x;
  const int tid = threadIdx.x;
  const float* sb = scores + (size_t)b * N_;
  const int*   mb = mask   + (size_t)b * N_;

  float mx = -3.402823466e38f;
  float cnt = 0.0f;
  for (int n = tid; n < N_; n += 256) {
    if (mb[n]) { mx = fmaxf(mx, sb[n]); cnt += 1.0f; }
  }
  const float M = block_reduce_max(mx, red, tid);
  const float C = block_reduce_sum(cnt, red, tid);

  float s = 0.0f;
  for (int n = tid; n < N_; n += 256)
    if (mb[n]) s += __expf(sb[n] - M);
  const float S = block_reduce_sum(s, red, tid);
  const float inv = 1.0f / S;

  for (int n = tid; n < N_; n += 256)
    weights[(size_t)b * N_ + n] = mb[n] ? __expf(sb[n] - M) * inv : 0.0f;

  if (tid == 0) inv_valid[b] = 1.0f / fmaxf(C, 1e-8f);
}

// ---------------------------------------------------------------------------
// Kernel 3a: partial pooling over an N-chunk of 256 rows.
// grid = B * NCHUNK * 2 = 1024 blocks (8192 waves -> enough loads in flight
// to saturate HBM). Each block: fixed b, one 256-row chunk, one 256-wide
// half of D. Softmax weights + mask staged in LDS once per block.
// ---------------------------------------------------------------------------
__global__ __launch_bounds__(256) void pool_partial_kernel(
    const float* __restrict__ x, const int* __restrict__ mask,
    const float* __restrict__ weights,
    float* __restrict__ attn_part, float* __restrict__ max_part,
    float* __restrict__ sum_part)
{
  __shared__ float wsh[CHUNK_];
  __shared__ int   msh[CHUNK_];

  const int bid   = blockIdx.x;
  const int dhalf = bid & 1;
  const int chunk = (bid >> 1) & (NCHUNK_ - 1);
  const int b     = bid >> 5;           // / (2 * NCHUNK_)
  const int tid   = threadIdx.x;
  const int d     = dhalf * 256 + tid;
  const int n0    = chunk * CHUNK_;

  wsh[tid] = weights[(size_t)b * N_ + n0 + tid];
  msh[tid] = mask[(size_t)b * N_ + n0 + tid];
  __syncthreads();

  const float* xb = x + ((size_t)b * N_ + n0) * D_ + d;
  float acc = 0.0f, mx = -3.402823466e38f, sm = 0.0f;
#pragma unroll 4
  for (int n = 0; n < CHUNK_; ++n) {
    const float v = __builtin_nontemporal_load(xb + (size_t)n * D_);
    acc = fmaf(wsh[n], v, acc);         // weights are 0 at masked positions
    if (msh[n]) { mx = fmaxf(mx, v); sm += v; }
  }
  const size_t o = ((size_t)b * NCHUNK_ + chunk) * D_ + d;
  attn_part[o] = acc;
  max_part[o]  = mx;
  sum_part[o]  = sm;
}

// ---------------------------------------------------------------------------
// Kernel 3b: fold the NCHUNK partials -> attn_p / max_p / mean_p.
// Deterministic (fixed order), grid = B*D/256 = 64 blocks.
// ---------------------------------------------------------------------------
__global__ __launch_bounds__(256) void pool_reduce_kernel(
    const float* __restrict__ attn_part, const float* __restrict__ max_part,
    const float* __restrict__ sum_part, const float* __restrict__ inv_valid,
    float* __restrict__ attn_p, float* __restrict__ max_p,
    float* __restrict__ mean_p)
{
  const int gid = blockIdx.x * 256 + threadIdx.x;  // 0 .. B*D-1
  const int b = gid / D_;
  const int d = gid % D_;
  float acc = 0.0f, mx = -3.402823466e38f, sm = 0.0f;
#pragma unroll
  for (int c = 0; c < NCHUNK_; ++c) {
    const size_t o = ((size_t)b * NCHUNK_ + c) * D_ + d;
    acc += attn_part[o];
    mx = fmaxf(mx, max_part[o]);
    sm += sum_part[o];
  }
  attn_p[gid] = acc;
  max_p[gid]  = mx;                     // >=1 valid row guaranteed -> finite
  mean_p[gid] = sm * inv_valid[b];
}

// ---------------------------------------------------------------------------
// LayerNorm over a 512-wide shared buffer (256 threads, 2 elems/thread).
// ---------------------------------------------------------------------------
__device__ __forceinline__ void ln_512(float* buf, const float* __restrict__ g,
                                       const float* __restrict__ bta,
                                       float* red, int tid)
{
  const float S = block_reduce_sum(buf[tid] + buf[tid + 256], red, tid);
  const float mean = S * (1.0f / 512.0f);
  const float v0 = buf[tid] - mean;
  const float v1 = buf[tid + 256] - mean;
  const float Q = block_reduce_sum(v0 * v0 + v1 * v1, red, tid);
  const float rstd = rsqrtf(Q * (1.0f / 512.0f) + 1e-5f);
  buf[tid]       = v0 * rstd * g[tid]       + bta[tid];
  buf[tid + 256] = v1 * rstd * g[tid + 256] + bta[tid + 256];
  __syncthreads();
}

// ---------------------------------------------------------------------------
// Kernel 4: gate MLP + LN + GELU + 3-way softmax gate + combine + out proj +
// LN. One block per batch row (32 rows total, negligible FLOPs).
// ---------------------------------------------------------------------------
__global__ __launch_bounds__(256) void head_kernel(
    const float* __restrict__ attn_p, const float* __restrict__ max_p,
    const float* __restrict__ mean_p,
    const float* __restrict__ gw1, const float* __restrict__ gb1,
    const float* __restrict__ gln_g, const float* __restrict__ gln_b,
    const float* __restrict__ gw2, const float* __restrict__ gb2,
    const float* __restrict__ ow, const float* __restrict__ ob,
    const float* __restrict__ oln_g, const float* __restrict__ oln_b,
    float* __restrict__ out)
{
  __shared__ float comb[3 * D_];
  __shared__ float gbuf[D_];
  __shared__ float obuf[OUT_];
  __shared__ float red[256];
  __shared__ float gates[3];

  const int b = blockIdx.x;
  const int tid = threadIdx.x;

  for (int i = tid; i < D_; i += 256) {
    comb[i]          = attn_p[b * D_ + i];
    comb[D_ + i]     = max_p[b * D_ + i];
    comb[2 * D_ + i] = mean_p[b * D_ + i];
  }
  __syncthreads();

  // g = combined @ gate_w1 + gate_b1   (1536 x 512)
  for (int d = tid; d < D_; d += 256) {
    float acc = gb1[d];
    for (int k = 0; k < 3 * D_; ++k)
      acc = fmaf(comb[k], gw1[(size_t)k * D_ + d], acc);
    gbuf[d] = acc;
  }
  __syncthreads();

  ln_512(gbuf, gln_g, gln_b, red, tid);

  // exact GELU
  for (int d = tid; d < D_; d += 256) {
    const float v = gbuf[d];
    gbuf[d] = 0.5f * v * (1.0f + erff(v * 0.70710678118654752f));
  }
  __syncthreads();

  // gates = softmax(g @ gate_w2 + gate_b2)  (3 logits)
  float p0 = 0.0f, p1 = 0.0f, p2 = 0.0f;
  for (int d = tid; d < D_; d += 256) {
    const float v = gbuf[d];
    p0 = fmaf(v, gw2[d * 3 + 0], p0);
    p1 = fmaf(v, gw2[d * 3 + 1], p1);
    p2 = fmaf(v, gw2[d * 3 + 2], p2);
  }
  const float s0 = block_reduce_sum(p0, red, tid);
  const float s1 = block_reduce_sum(p1, red, tid);
  const float s2 = block_reduce_sum(p2, red, tid);
  if (tid == 0) {
    const float l0 = s0 + gb2[0], l1 = s1 + gb2[1], l2 = s2 + gb2[2];
    const float m = fmaxf(l0, fmaxf(l1, l2));
    const float e0 = __expf(l0 - m), e1 = __expf(l1 - m), e2 = __expf(l2 - m);
    const float inv = 1.0f / (e0 + e1 + e2);
    gates[0] = e0 * inv; gates[1] = e1 * inv; gates[2] = e2 * inv;
  }
  __syncthreads();

  // weighted combine (overwrites gbuf; gbuf no longer needed)
  for (int d = tid; d < D_; d += 256)
    gbuf[d] = comb[d] * gates[0] + comb[D_ + d] * gates[1] +
              comb[2 * D_ + d] * gates[2];
  __syncthreads();

  // out = weighted @ out_w + out_b  (512 x 512)
  for (int o = tid; o < OUT_; o += 256) {
    float acc = ob[o];
    for (int k = 0; k < D_; ++k)
      acc = fmaf(gbuf[k], ow[(size_t)k * OUT_ + o], acc);
    obuf[o] = acc;
  }
  __syncthreads();

  ln_512(obuf, oln_g, oln_b, red, tid);

  out[(size_t)b * OUT_ + tid]       = obuf[tid];
  out[(size_t)b * OUT_ + tid + 256] = obuf[tid + 256];
}

// ---------------------------------------------------------------------------
extern "C" void kernel_launch(void* const* d_in, const int* in_sizes, int n_in,
                              void* d_out, int out_size, void* d_ws, size_t ws_size,
                              hipStream_t stream) {
  const float* x        = (const float*)d_in[0];
  const int*   mask     = (const int*)d_in[1];
  const float* attn_w1  = (const float*)d_in[2];
  const float* attn_b1  = (const float*)d_in[3];
  const float* attn_w2  = (const float*)d_in[4];
  const float* attn_b2  = (const float*)d_in[5];
  const float* gate_w1  = (const float*)d_in[6];
  const float* gate_b1  = (const float*)d_in[7];
  const float* gate_lng = (const float*)d_in[8];
  const float* gate_lnb = (const float*)d_in[9];
  const float* gate_w2  = (const float*)d_in[10];
  const float* gate_b2  = (const float*)d_in[11];
  const float* out_w    = (const float*)d_in[12];
  const float* out_b    = (const float*)d_in[13];
  const float* out_lng  = (const float*)d_in[14];
  const float* out_lnb  = (const float*)d_in[15];
  float* out = (float*)d_out;

  float* ws        = (float*)d_ws;
  float* scores    = ws;                                    // B*N
  float* weights   = scores    + (size_t)B_ * N_;           // B*N
  float* attn_part = weights   + (size_t)B_ * N_;           // B*NCHUNK*D
  float* max_part  = attn_part + (size_t)B_ * NCHUNK_ * D_; // B*NCHUNK*D
  float* sum_part  = max_part  + (size_t)B_ * NCHUNK_ * D_; // B*NCHUNK*D
  float* attn_p    = sum_part  + (size_t)B_ * NCHUNK_ * D_; // B*D
  float* max_p     = attn_p    + (size_t)B_ * D_;
  float* mean_p    = max_p     + (size_t)B_ * D_;
  float* inv_valid = mean_p    + (size_t)B_ * D_;           // B

  attn_score_kernel<<<(B_ * N_) / 256, 256, 0, stream>>>(
      x, attn_w1, attn_b1, attn_w2, attn_b2, scores);
  softmax_mask_kernel<<<B_, 256, 0, stream>>>(scores, mask, weights, inv_valid);
  pool_partial_kernel<<<B_ * NCHUNK_ * 2, 256, 0, stream>>>(
      x, mask, weights, attn_part, max_part, sum_part);
  pool_reduce_kernel<<<(B_ * D_) / 256, 256, 0, stream>>>(
      attn_part, max_part, sum_part, inv_valid, attn_p, max_p, mean_p);
  head_kernel<<<B_, 256, 0, stream>>>(
      attn_p, max_p, mean_p, gate_w1, gate_b1, gate_lng, gate_lnb,
      gate_w2, gate_b2, out_w, out_b, out_lng, out_lnb, out);
}